// Top2Gating_16621523435697
// MI455X (gfx1250) — compile-verified
//
#include <hip/hip_runtime.h>
#include <hip/hip_bf16.h>
#include <math.h>

// Problem constants (match reference)
constexpr int S = 8192;   // tokens
constexpr int D = 2048;   // model dim
constexpr int E = 16;     // experts
constexpr int C = 1024;   // capacity = 2*S/E
constexpr int TPB = 256;  // tokens per block for routing passes
constexpr int NB = S / TPB; // 32 blocks

// Workspace layout (byte offsets)
constexpr size_t LOGITS_OFF = 0;                          // float[S*E]   = 512 KB
constexpr size_t IDX1_OFF   = LOGITS_OFF + (size_t)S*E*4; // int[S]
constexpr size_t IDX2_OFF   = IDX1_OFF + (size_t)S*4;
constexpr size_t G1_OFF     = IDX2_OFF + (size_t)S*4;     // float[S]
constexpr size_t G2_OFF     = G1_OFF   + (size_t)S*4;
constexpr size_t CNT1_OFF   = G2_OFF   + (size_t)S*4;     // int[NB*E]
constexpr size_t CNT2_OFF   = CNT1_OFF + (size_t)NB*E*4;
constexpr size_t GSUM_OFF   = CNT2_OFF + (size_t)NB*E*4;  // float[NB*E]
constexpr size_t OFF1_OFF   = GSUM_OFF + (size_t)NB*E*4;  // int[NB*E]
constexpr size_t OFF2_OFF   = OFF1_OFF + (size_t)NB*E*4;
constexpr size_t TOT1_OFF   = OFF2_OFF + (size_t)NB*E*4;  // int[E]

typedef __attribute__((ext_vector_type(2))) float v2f;
typedef __attribute__((ext_vector_type(8))) float v8f;

// ---------------------------------------------------------------------------
// K0: zero-fill the ~1 GB output (the true bandwidth bottleneck). b128 stores.
// ---------------------------------------------------------------------------
__global__ __launch_bounds__(256) void zero_out_kernel(float* __restrict__ out, long long n) {
    long long n4 = n >> 2;
    float4* p4 = (float4*)out;
    long long i = (long long)blockIdx.x * blockDim.x + threadIdx.x;
    long long stride = (long long)gridDim.x * blockDim.x;
    float4 z; z.x = 0.f; z.y = 0.f; z.z = 0.f; z.w = 0.f;
    for (; i < n4; i += stride) p4[i] = z;
    if (blockIdx.x == 0 && threadIdx.x == 0) {
        for (long long k = n4 << 2; k < n; ++k) out[k] = 0.f;
    }
}

// ---------------------------------------------------------------------------
// K1: logits = x @ wg via V_WMMA_F32_16X16X4_F32 (full f32 precision).
// 8 waves per block; wave w owns M-tile (blockIdx*8 + w)*16, all 16 experts.
// A 16x4: lane L -> row L&15, K-pair 2*(L>>4) (contiguous 8B load).
// B 4x16: lane L -> col L&15, same K-pair (two b32 loads, 64B apart).
// ---------------------------------------------------------------------------
__global__ __launch_bounds__(256) void gemm_logits_wmma(const float* __restrict__ x,
                                                        const float* __restrict__ wg,
                                                        float* __restrict__ logits) {
    const int lane  = threadIdx.x & 31;
    const int wave  = threadIdx.x >> 5;
    const int rc    = lane & 15;      // row of A / col of B
    const int khalf = lane >> 4;      // 0 or 1
    const int mbase = (blockIdx.x * 8 + wave) * 16;

    const float* xrow = x + (size_t)(mbase + rc) * D + 2 * khalf;

    v8f c = {};
    for (int k0 = 0; k0 < D; k0 += 4) {
        v2f a = *(const v2f*)(xrow + k0);                 // 8B-aligned b64 load
        const float* wp = wg + (size_t)(k0 + 2 * khalf) * E + rc;
        v2f b;
        b.x = wp[0];
        b.y = wp[E];
        c = __builtin_amdgcn_wmma_f32_16x16x4_f32(
                /*neg_a=*/false, a, /*neg_b=*/false, b,
                /*c_mod=*/(short)0, c, /*reuse_a=*/false, /*reuse_b=*/false);
    }
    // C/D layout: lane L, VGPR r -> (M = r + 8*khalf, N = rc)
    #pragma unroll
    for (int r = 0; r < 8; ++r) {
        int m = mbase + r + 8 * khalf;
        logits[(size_t)m * E + rc] = c[r];
    }
}

// ---------------------------------------------------------------------------
// K2: per-token softmax + top-2 + per-block expert histograms and gate sums.
// ---------------------------------------------------------------------------
__global__ __launch_bounds__(TPB) void token_pass1(const float* __restrict__ logits,
                                                   int* __restrict__ idx1, int* __restrict__ idx2,
                                                   float* __restrict__ g1, float* __restrict__ g2,
                                                   int* __restrict__ cnt1, int* __restrict__ cnt2,
                                                   float* __restrict__ gsum) {
    __shared__ int   c1[E];
    __shared__ int   c2[E];
    __shared__ float gs[E];
    const int t = threadIdx.x;
    const int s = blockIdx.x * TPB + t;
    if (t < E) { c1[t] = 0; c2[t] = 0; gs[t] = 0.f; }
    __syncthreads();

    float l[E];
    const float* lp = logits + (size_t)s * E;
    float m = -INFINITY;
    #pragma unroll
    for (int e = 0; e < E; ++e) { l[e] = lp[e]; m = fmaxf(m, l[e]); }
    float ge[E]; float sum = 0.f;
    #pragma unroll
    for (int e = 0; e < E; ++e) { ge[e] = __expf(l[e] - m); sum += ge[e]; }
    const float inv = 1.0f / sum;

    // argmax (first occurrence, matching jnp.argmax)
    int e1 = 0; float b1 = l[0];
    #pragma unroll
    for (int e = 1; e < E; ++e) if (l[e] > b1) { b1 = l[e]; e1 = e; }
    int e2 = (e1 == 0) ? 1 : 0; float b2 = -INFINITY;
    #pragma unroll
    for (int e = 0; e < E; ++e) if (e != e1 && l[e] > b2) { b2 = l[e]; e2 = e; }

    idx1[s] = e1; idx2[s] = e2;
    g1[s] = ge[e1] * inv;
    g2[s] = ge[e2] * inv;

    atomicAdd(&c1[e1], 1);
    atomicAdd(&c2[e2], 1);
    #pragma unroll
    for (int e = 0; e < E; ++e) atomicAdd(&gs[e], ge[e] * inv);
    __syncthreads();
    if (t < E) {
        cnt1[blockIdx.x * E + t] = c1[t];
        cnt2[blockIdx.x * E + t] = c2[t];
        gsum[blockIdx.x * E + t] = gs[t];
    }
}

// ---------------------------------------------------------------------------
// K3: tiny cross-block exclusive scan per expert + l_aux. Single wave32.
// ---------------------------------------------------------------------------
__global__ __launch_bounds__(32) void scan_blocks(const int* __restrict__ cnt1,
                                                  const int* __restrict__ cnt2,
                                                  const float* __restrict__ gsum,
                                                  int* __restrict__ off1, int* __restrict__ off2,
                                                  int* __restrict__ tot1,
                                                  float* __restrict__ out /* d_out; [0]=l_aux */) {
    __shared__ float part[E];
    const int e = threadIdx.x;
    if (e < E) {
        int r1 = 0, r2 = 0; float gsr = 0.f;
        for (int b = 0; b < NB; ++b) {
            off1[b * E + e] = r1; r1 += cnt1[b * E + e];
            off2[b * E + e] = r2; r2 += cnt2[b * E + e];
            gsr += gsum[b * E + e];
        }
        tot1[e] = r1;                          // pre-capacity count of top-1 per expert
        const float me = gsr / (float)S;       // mean gate
        const float ce = (float)r1 / (float)S; // mean mask1
        part[e] = me * ce;
    }
    __syncthreads();
    if (e == 0) {
        float a = 0.f;
        for (int i = 0; i < E; ++i) a += part[i];
        out[0] = a / (float)E;                 // l_aux
    }
}

// ---------------------------------------------------------------------------
// K4: in-block stable ranks -> locations, capacity drop, renormalize, scatter.
// ---------------------------------------------------------------------------
__global__ __launch_bounds__(TPB) void token_pass2(const int* __restrict__ idx1,
                                                   const int* __restrict__ idx2,
                                                   const float* __restrict__ g1,
                                                   const float* __restrict__ g2,
                                                   const int* __restrict__ off1,
                                                   const int* __restrict__ off2,
                                                   const int* __restrict__ tot1,
                                                   float* __restrict__ out) {
    __shared__ int ex1[TPB];
    __shared__ int ex2[TPB];
    const int t = threadIdx.x;
    const int s = blockIdx.x * TPB + t;
    const int e1 = idx1[s];
    const int e2 = idx2[s];
    ex1[t] = e1; ex2[t] = e2;
    __syncthreads();

    int r1 = 0, r2 = 0;
    for (int j = 0; j < t; ++j) {
        r1 += (ex1[j] == e1);
        r2 += (ex2[j] == e2);
    }
    const int loc1 = off1[blockIdx.x * E + e1] + r1;
    const int loc2 = off2[blockIdx.x * E + e2] + r2 + tot1[e2]; // + sum(mask1) offset
    const bool k1 = loc1 < C;
    const bool k2 = loc2 < C;

    const float a = k1 ? g1[s] : 0.f;
    const float b = k2 ? g2[s] : 0.f;
    const float denom = fmaxf(a + b, 1.1920929e-7f); // FLT_EPSILON clip
    const float invd = 1.0f / denom;

    float* comb = out + 1;
    float* disp = out + 1 + (size_t)S * E * C;
    const size_t base = (size_t)s * E * C;
    if (k1) {
        size_t i = base + (size_t)e1 * C + loc1;
        comb[i] = a * invd;
        disp[i] = 1.0f;
    }
    if (k2) {
        size_t i = base + (size_t)e2 * C + loc2;
        comb[i] = b * invd;
        disp[i] = 1.0f;
    }
}

// ---------------------------------------------------------------------------
extern "C" void kernel_launch(void* const* d_in, const int* in_sizes, int n_in,
                              void* d_out, int out_size, void* d_ws, size_t ws_size,
                              hipStream_t stream) {
    const float* x  = (const float*)d_in[0];   // [S, D]
    const float* wg = (const float*)d_in[1];   // [D, E]
    float* out = (float*)d_out;
    char* ws = (char*)d_ws;

    float* logits = (float*)(ws + LOGITS_OFF);
    int*   idx1   = (int*)  (ws + IDX1_OFF);
    int*   idx2   = (int*)  (ws + IDX2_OFF);
    float* g1     = (float*)(ws + G1_OFF);
    float* g2     = (float*)(ws + G2_OFF);
    int*   cnt1   = (int*)  (ws + CNT1_OFF);
    int*   cnt2   = (int*)  (ws + CNT2_OFF);
    float* gsum   = (float*)(ws + GSUM_OFF);
    int*   off1   = (int*)  (ws + OFF1_OFF);
    int*   off2   = (int*)  (ws + OFF2_OFF);
    int*   tot1   = (int*)  (ws + TOT1_OFF);

    // 1) zero-fill entire output (dominant cost: ~1 GB of b128 stores)
    zero_out_kernel<<<4096, 256, 0, stream>>>(out, (long long)out_size);
    // 2) logits via f32 WMMA (512 blocks worth of 16x16 tiles; 8 waves/block)
    gemm_logits_wmma<<<S / (16 * 8), 256, 0, stream>>>(x, wg, logits);
    // 3) per-token top-2 + block histograms
    token_pass1<<<NB, TPB, 0, stream>>>(logits, idx1, idx2, g1, g2, cnt1, cnt2, gsum);
    // 4) cross-block scan + l_aux
    scan_blocks<<<1, 32, 0, stream>>>(cnt1, cnt2, gsum, off1, off2, tot1, out);
    // 5) locations, capacity, normalize, sparse scatter
    token_pass2<<<NB, TPB, 0, stream>>>(idx1, idx2, g1, g2, off1, off2, tot1, out);
}